// TypedCoords2Volume_79585743995278
// MI455X (gfx1250) — compile-verified
//
#include <hip/hip_runtime.h>
#include <hip/hip_bf16.h>

// ---------------------------------------------------------------------------
// TypedCoords2Volume for MI455X (gfx1250).
//
// Cost model: output = 4*11*120^3 floats = 304 MB -> ~13us at 23.3 TB/s HBM.
// Compute (2.8M gaussian evals + float atomics) is negligible, so the kernel
// is: (1) wide-store zero fill, (2) scatter with global_atomic_add_f32.
// The r^2 = |off - frac|^2 cross-terms for a 16-atom x 16-offset tile are
// computed with V_WMMA_F32_16X16X4_F32:  D = (-2*d) x off^T + (|d|^2+|off|^2)
// which yields r^2 directly in the accumulator.
// ---------------------------------------------------------------------------

#define BOX        120
#define BOX3       (BOX * BOX * BOX)          // 1,728,000
#define BATCH      4
#define NTYPES     11
#define NMAPS      (BATCH * NTYPES)           // 44
#define MAXA       512
#define NOFF       125                        // 5^3 neighbour offsets
#define ATOMS_PER_BLOCK 64                    // 4 wave32s x 16 atoms
#define CHUNKS     (MAXA / ATOMS_PER_BLOCK)   // 8

typedef __attribute__((ext_vector_type(2))) float v2f;
typedef __attribute__((ext_vector_type(8))) float v8f;

// ---------------------------------------------------------------------------
// Kernel 1: zero the 304 MB output with 16B stores (global_store_b128).
// ---------------------------------------------------------------------------
__global__ __launch_bounds__(256)
void zero_kernel(float* __restrict__ out, long n4) {
    long i = (long)blockIdx.x * blockDim.x + threadIdx.x;
    if (i < n4) {
        float4 z = make_float4(0.f, 0.f, 0.f, 0.f);
        *reinterpret_cast<float4*>(out + i * 4) = z;
    }
}

// ---------------------------------------------------------------------------
// Kernel 2: gaussian splat. One block = one (bt) map x 64 atoms.
// Each wave32 handles 16 atoms; 8 WMMA tiles cover the 125 offsets.
// ---------------------------------------------------------------------------
__global__ __launch_bounds__(128)
void splat_kernel(const float* __restrict__ coords,
                  const int*   __restrict__ num_atoms,
                  float*       __restrict__ out) {
    __shared__ float s_dx[ATOMS_PER_BLOCK];
    __shared__ float s_dy[ATOMS_PER_BLOCK];
    __shared__ float s_dz[ATOMS_PER_BLOCK];
    __shared__ float s_q [ATOMS_PER_BLOCK];
    __shared__ int   s_cx[ATOMS_PER_BLOCK];
    __shared__ int   s_cy[ATOMS_PER_BLOCK];
    __shared__ int   s_cz[ATOMS_PER_BLOCK];

    const int tid   = threadIdx.x;
    const int bt    = blockIdx.x / CHUNKS;            // which of 44 maps
    const int chunk = blockIdx.x % CHUNKS;
    const int abase = chunk * ATOMS_PER_BLOCK;        // first atom of block

    const float* cb = coords + (size_t)bt * (3 * MAXA);

    // Phase 1: stage per-atom cell / fractional position / |d|^2 in LDS.
    if (tid < ATOMS_PER_BLOCK) {
        int a = abase + tid;
        float x = cb[3 * a + 0];
        float y = cb[3 * a + 1];
        float z = cb[3 * a + 2];
        float fx = floorf(x), fy = floorf(y), fz = floorf(z);
        float dx = x - fx, dy = y - fy, dz = z - fz;
        s_dx[tid] = dx; s_dy[tid] = dy; s_dz[tid] = dz;
        s_q[tid]  = dx * dx + dy * dy + dz * dz;
        s_cx[tid] = (int)fx; s_cy[tid] = (int)fy; s_cz[tid] = (int)fz;
    }
    __syncthreads();

    const int na   = num_atoms[bt];
    const int lane = tid & 31;
    const int wave = tid >> 5;
    const int wab  = wave * 16;           // this wave's atom base within LDS
    const int half = lane >> 4;           // 0: lanes 0-15, 1: lanes 16-31
    const int nl   = lane & 15;           // column index within tile

    // A operand (16x4 f32, rows = -2*d padded with 0):
    //   lane l row M=l%16; VGPR0/1 carry K = 2*half + {0,1}.
    v2f Aop;
    {
        int arow = wab + nl;
        if (half == 0) { Aop.x = -2.f * s_dx[arow]; Aop.y = -2.f * s_dy[arow]; }
        else           { Aop.x = -2.f * s_dz[arow]; Aop.y = 0.f; }
    }

    // |d|^2 for this lane's 8 accumulator rows (M = i + 8*half), loaded once.
    float qreg[8];
    const int mb = wab + 8 * half;
#pragma unroll
    for (int i = 0; i < 8; ++i) qreg[i] = s_q[mb + i];

    float* outbt = out + (size_t)bt * BOX3;

    // 8 tiles of 16 offsets cover the 125 neighbour offsets (last 3 masked).
    for (int t = 0; t < 8; ++t) {
        int n  = t * 16 + nl;                 // offset index 0..127
        int ko = (n < NOFF) ? n : (NOFF - 1); // clamp padding columns
        int ox = ko / 25 - 2;
        int oy = (ko / 5) % 5 - 2;
        int oz = ko % 5 - 2;

        // B operand (4x16 f32): lane l column N=l%16; K = 2*half + {0,1}.
        v2f Bop;
        if (half == 0) { Bop.x = (float)ox; Bop.y = (float)oy; }
        else           { Bop.x = (float)oz; Bop.y = 0.f; }

        float S = (float)(ox * ox + oy * oy + oz * oz);

        // C preloaded with |o|^2 + |d|^2 so D comes out as r^2 directly.
        v8f C;
#pragma unroll
        for (int i = 0; i < 8; ++i) C[i] = qreg[i] + S;

        // D = A x B + C  (uniform control flow: EXEC all ones here)
        v8f R = __builtin_amdgcn_wmma_f32_16x16x4_f32(
            /*neg_a=*/false, Aop, /*neg_b=*/false, Bop,
            /*c_mod=*/(short)0, C, /*reuse_a=*/false, /*reuse_b=*/false);

        // Scatter this lane's 8 (atom, offset) results.
        if (n < NOFF) {
#pragma unroll
            for (int i = 0; i < 8; ++i) {
                int m  = mb + i;          // atom index within block's LDS tile
                int ag = abase + m;       // atom index within this map
                if (ag < na) {
                    int gx = s_cx[m] + ox;
                    int gy = s_cy[m] + oy;
                    int gz = s_cz[m] + oz;
                    if ((unsigned)gx < (unsigned)BOX &&
                        (unsigned)gy < (unsigned)BOX &&
                        (unsigned)gz < (unsigned)BOX) {
                        float val = __expf(-R[i]);
                        atomicAdd(outbt + ((size_t)gx * BOX + gy) * BOX + gz, val);
                    }
                }
            }
        }
    }
}

extern "C" void kernel_launch(void* const* d_in, const int* in_sizes, int n_in,
                              void* d_out, int out_size, void* d_ws, size_t ws_size,
                              hipStream_t stream) {
    const float* coords    = (const float*)d_in[0];   // [4, 11, 3*512] f32
    const int*   num_atoms = (const int*)d_in[1];     // [4, 11] int
    float*       out       = (float*)d_out;           // [4,11,120,120,120] f32

    // Zero-fill 304 MB with 16B stores.
    long n4 = (long)out_size / 4;                     // 19,008,000 float4s
    int  zgrid = (int)((n4 + 255) / 256);
    zero_kernel<<<zgrid, 256, 0, stream>>>(out, n4);

    // Gaussian splat: 44 maps x 8 chunks of 64 atoms.
    splat_kernel<<<NMAPS * CHUNKS, 128, 0, stream>>>(coords, num_atoms, out);
}